// Circuit_67473936220746
// MI455X (gfx1250) — compile-verified
//
#include <hip/hip_runtime.h>
#include <cstdint>

typedef __attribute__((ext_vector_type(2))) float v2f;
typedef __attribute__((ext_vector_type(8))) float v8f;

#define N_STATE (1u << 23)   // elements per channel
#define UDIM 128
#define ROWS 32              // m-rows per block (2 m-tiles per wave)

struct GateMap { int pos_a[7]; int pos_m[16]; };

// ---------------------------------------------------------------------------
// One gate: out[ch][scatter(m)|scatter_a(k)], Karatsuba complex matmul vs
// Ur/Ui (128x128 row-major, out[m,k] = sum_a U[k,a]*s[m,a]).
// Channel swap on store: ch0 <- imag, ch1 <- real (faithful to reference).
// Each block: 32 rows x 128 cols; 8 waves, each wave one 16-col tile over
// two 16-row tiles (B fragments reused across both -> fewer vmem per WMMA).
// ---------------------------------------------------------------------------
__global__ __launch_bounds__(256)
void Circuit_gate_wmma(const float* in, float* out,
                       const float* Ur, const float* Ui,
                       GateMap gm, int contig) {
  __shared__ float lds_sr[ROWS][132];  // stride 132: 132%64=4 -> conflict-free
  __shared__ float lds_si[ROWS][132];
  __shared__ int   lutA[128];
  __shared__ int   lutM[ROWS];

  const int tid   = threadIdx.x;
  const int mbase = blockIdx.x << 5;

  // --- permutation scatter LUTs (disjoint bit fields => OR == ADD) ---
  if (tid < 128) {
    int v = 0;
#pragma unroll
    for (int j = 0; j < 7; ++j) v |= ((tid >> j) & 1) << gm.pos_a[j];
    lutA[tid] = v;
  } else if (tid < 128 + ROWS) {
    int m = mbase + (tid - 128);
    int v = 0;
#pragma unroll
    for (int b = 0; b < 16; ++b) v |= ((m >> b) & 1) << gm.pos_m[b];
    lutM[tid - 128] = v;
  }
  __syncthreads();

  // --- async gather: 32x128 tile, both channels, global -> LDS direct ---
  if (contig) {
    // a-axis is the low 7 address bits: 16B-contiguous -> async b128
#pragma unroll
    for (int i = 0; i < 4; ++i) {
      int e = tid + i * 256;            // 0..1023 quad-slots
      int r = e >> 5;
      int a = (e & 31) << 2;
      int idx = lutM[r] | lutA[a];
      const float* gr = in + idx;
      const float* gi = in + N_STATE + idx;
      uint32_t dr = (uint32_t)(uintptr_t)&lds_sr[r][a];
      uint32_t di = (uint32_t)(uintptr_t)&lds_si[r][a];
      asm volatile("global_load_async_to_lds_b128 %0, %1, off"
                   :: "v"(dr), "v"(gr) : "memory");
      asm volatile("global_load_async_to_lds_b128 %0, %1, off"
                   :: "v"(di), "v"(gi) : "memory");
    }
  } else {
    // strided a-axis (gate 3): per-element async b32
#pragma unroll
    for (int i = 0; i < 16; ++i) {
      int e = tid + i * 256;            // 0..4095
      int r = e >> 7;
      int a = e & 127;
      int idx = lutM[r] | lutA[a];
      const float* gr = in + idx;
      const float* gi = in + N_STATE + idx;
      uint32_t dr = (uint32_t)(uintptr_t)&lds_sr[r][a];
      uint32_t di = (uint32_t)(uintptr_t)&lds_si[r][a];
      asm volatile("global_load_async_to_lds_b32 %0, %1, off"
                   :: "v"(dr), "v"(gr) : "memory");
      asm volatile("global_load_async_to_lds_b32 %0, %1, off"
                   :: "v"(di), "v"(gi) : "memory");
    }
  }
  asm volatile("s_wait_asynccnt 0x0" ::: "memory");
  __syncthreads();

  const int lane = tid & 31;
  const int wave = tid >> 5;            // 0..7 -> k-tile
  const int k0   = wave << 4;
  const int mrow = lane & 15;           // A: M index / B,D: N index
  const int hi   = lane >> 4;           // K-pair select (A/B), +8 row (D)

  v8f aR0 = {}, aI0 = {}, aS0 = {};     // rows 0..15
  v8f aR1 = {}, aI1 = {}, aS1 = {};     // rows 16..31

  // B fragment: B[a][k] = U[k*128+a]; lane k = k0+mrow, K-pair a = 4s+2*hi.
  const float* urp = Ur + (size_t)(k0 + mrow) * UDIM + 2 * hi;
  const float* uip = Ui + (size_t)(k0 + mrow) * UDIM + 2 * hi;

#pragma unroll 2
  for (int s = 0; s < 32; ++s) {
    const int aoff = 4 * s + 2 * hi;
    v2f br = *(const v2f*)(urp + 4 * s);
    v2f bi = *(const v2f*)(uip + 4 * s);
    v2f bs = br + bi;
    v2f ar0 = *(const v2f*)&lds_sr[mrow][aoff];
    v2f ai0 = *(const v2f*)&lds_si[mrow][aoff];
    v2f ar1 = *(const v2f*)&lds_sr[mrow + 16][aoff];
    v2f ai1 = *(const v2f*)&lds_si[mrow + 16][aoff];
    v2f as0 = ar0 + ai0;
    v2f as1 = ar1 + ai1;
    aR0 = __builtin_amdgcn_wmma_f32_16x16x4_f32(false, ar0, false, br, (short)0, aR0, false, false);
    aI0 = __builtin_amdgcn_wmma_f32_16x16x4_f32(false, ai0, false, bi, (short)0, aI0, false, false);
    aS0 = __builtin_amdgcn_wmma_f32_16x16x4_f32(false, as0, false, bs, (short)0, aS0, false, false);
    aR1 = __builtin_amdgcn_wmma_f32_16x16x4_f32(false, ar1, false, br, (short)0, aR1, false, false);
    aI1 = __builtin_amdgcn_wmma_f32_16x16x4_f32(false, ai1, false, bi, (short)0, aI1, false, false);
    aS1 = __builtin_amdgcn_wmma_f32_16x16x4_f32(false, as1, false, bs, (short)0, aS1, false, false);
  }

  // --- epilogue: D layout: VGPR v -> M = v + 8*hi, N = mrow ---
  const int col  = k0 + mrow;
  const int offA = lutA[col];
#pragma unroll
  for (int v = 0; v < 8; ++v) {
    int r = v + 8 * hi;
    {
      float wr = aR0[v], wi = aI0[v], ss = aS0[v];
      int idx = lutM[r] | offA;
      out[idx]           = ss - wr - wi;  // ch0 <- imag (reference swap)
      out[N_STATE + idx] = wr - wi;       // ch1 <- real
    }
    {
      float wr = aR1[v], wi = aI1[v], ss = aS1[v];
      int idx = lutM[r + 16] | offA;
      out[idx]           = ss - wr - wi;
      out[N_STATE + idx] = wr - wi;
    }
  }
}

// ---------------------------------------------------------------------------
// Host-side permutation math. Reference: perm = [ch] + idcs + rem(ascending),
// then reshape(2,-1,128): the 128-dim is the LAST 7 axes = rem[9..15];
// M-dim axes (MSB->LSB) = idcs[0..6], rem[0..8]. Qubit q <-> addr bit (22-q).
// ---------------------------------------------------------------------------
static GateMap make_map(const int* idcs) {
  bool used[23] = {};
  for (int i = 0; i < 7; ++i) used[idcs[i]] = true;
  int rem[16], n = 0;
  for (int q = 0; q < 23; ++q)
    if (!used[q]) rem[n++] = q;
  GateMap g;
  for (int j = 0; j < 7; ++j)  g.pos_a[j]      = 22 - rem[15 - j]; // a bit j
  for (int i = 0; i < 7; ++i)  g.pos_m[15 - i] = 22 - idcs[i];     // m top bits
  for (int t = 0; t < 9; ++t)  g.pos_m[8 - t]  = 22 - rem[t];      // m low bits
  return g;
}

extern "C" void kernel_launch(void* const* d_in, const int* in_sizes, int n_in,
                              void* d_out, int out_size, void* d_ws, size_t ws_size,
                              hipStream_t stream) {
  (void)in_sizes; (void)n_in; (void)out_size; (void)d_ws; (void)ws_size;
  const float* state = (const float*)d_in[0];
  const float* U     = (const float*)d_in[1];
  float* out = (float*)d_out;

  static const int GATE_IDCS[4][7] = {
      {0, 1, 2, 3, 4, 5, 6},
      {7, 8, 9, 10, 11, 12, 13},
      {3, 5, 7, 9, 11, 13, 15},
      {16, 17, 18, 19, 20, 21, 22}};

  const int n_blocks = (1 << 16) / ROWS;  // 2048 blocks, 32 rows each

  for (int g = 0; g < 4; ++g) {
    GateMap gm = make_map(GATE_IDCS[g]);
    int contig = (gm.pos_a[0] == 0 && gm.pos_a[1] == 1) ? 1 : 0;
    const float* Ugr = U + (size_t)g * 2 * UDIM * UDIM;
    const float* Ugi = Ugr + UDIM * UDIM;
    const float* src = (g == 0) ? state : out;  // gates 1..3 run in place
    Circuit_gate_wmma<<<n_blocks, 256, 0, stream>>>(src, out, Ugr, Ugi, gm, contig);
  }
}